// GHMCLoss_35974646072039
// MI455X (gfx1250) — compile-verified
//
#include <hip/hip_runtime.h>
#include <stdint.h>

#define BINS   10
#define TILE   256          // float4 elements per tile per array
#define STRIDE 21           // per-thread LDS histogram stride (dwords), odd -> spreads banks
#define MAXGRID 1536

// ---- CDNA5 async global->LDS copy (16B per active lane), tracked by ASYNCcnt ----
__device__ __forceinline__ void async_copy16(uint32_t lds_addr, const void* gptr) {
    asm volatile("global_load_async_to_lds_b128 %0, %1, off"
                 :: "v"(lds_addr),
                    "v"((unsigned long long)(uintptr_t)gptr)
                 : "memory");
}
#define WAIT_ASYNC(N) asm volatile("s_wait_asynccnt %0" :: "i"(N) : "memory")

// ---- per-element math: bin index + unweighted BCE, accumulated into private LDS bins ----
__device__ __forceinline__ void ghmc_elem(float xv, float yv, unsigned* bins) {
    float ax  = fabsf(xv);
    float e   = __expf(-ax);                        // v_exp_f32 path, e in (0,1]
    float r   = __builtin_amdgcn_rcpf(1.0f + e);    // 1/(1+e)
    float sig = (xv >= 0.0f) ? r : (1.0f - r);      // sigmoid(x)
    float g   = fabsf(sig - yv);                    // in [0,1]
    int   idx = (int)(g * 10.0f);                   // trunc == floor for g>=0
    idx = idx > (BINS - 1) ? (BINS - 1) : idx;
    // bce = max(x,0) - x*y + log1p(exp(-|x|));  log(1+e) safe: e<=1, no cancellation
    float bce = fmaxf(xv, 0.0f) - xv * yv + __logf(1.0f + e);
    atomicAdd(&bins[idx], 1u);                       // ds_add_u32 (private region, no contention)
    atomicAdd((float*)&bins[BINS + idx], bce);       // ds_add_f32
}

__global__ __launch_bounds__(TILE)
void ghmc_main(const float* __restrict__ x, const float* __restrict__ y,
               unsigned* __restrict__ ws, int n, int n4, int ntiles) {
    __shared__ float4   sX[2][TILE];
    __shared__ float4   sY[2][TILE];
    __shared__ unsigned sHist[TILE * STRIDE];

    const int tid = threadIdx.x;
    unsigned* mybins = &sHist[tid * STRIDE];
#pragma unroll
    for (int i = 0; i < 2 * BINS; ++i) mybins[i] = 0u;

    const float4* x4 = (const float4*)x;
    const float4* y4 = (const float4*)y;

    // flat->LDS: low 32 bits of a flat shared-aperture address are the LDS byte address
    uint32_t ldsX[2] = { (uint32_t)(uintptr_t)&sX[0][tid], (uint32_t)(uintptr_t)&sX[1][tid] };
    uint32_t ldsY[2] = { (uint32_t)(uintptr_t)&sY[0][tid], (uint32_t)(uintptr_t)&sY[1][tid] };

    const int t0     = (int)blockIdx.x;
    const int stride = (int)gridDim.x;

    // prologue: stage first tile into buffer 0
    if (t0 < ntiles) {
        int i4 = t0 * TILE + tid;
        if (i4 < n4) {
            async_copy16(ldsX[0], x4 + i4);
            async_copy16(ldsY[0], y4 + i4);
        }
    }

    int buf = 0;
    for (int t = t0; t < ntiles; t += stride) {
        int nxt = t + stride;
        if (nxt < ntiles) {
            int j4 = nxt * TILE + tid;
            if (j4 < n4) {
                async_copy16(ldsX[buf ^ 1], x4 + j4);
                async_copy16(ldsY[buf ^ 1], y4 + j4);
            }
            WAIT_ASYNC(2);          // keep only the freshly issued stage in flight
        } else {
            WAIT_ASYNC(0);          // drain: last tile
        }
        int i4 = t * TILE + tid;
        if (i4 < n4) {
            float4 xv = sX[buf][tid];   // ds_load_b128
            float4 yv = sY[buf][tid];
            ghmc_elem(xv.x, yv.x, mybins);
            ghmc_elem(xv.y, yv.y, mybins);
            ghmc_elem(xv.z, yv.z, mybins);
            ghmc_elem(xv.w, yv.w, mybins);
        }
        buf ^= 1;
    }

    // scalar tail (n % 4), handled once by block 0 / thread 0 before the block reduce
    if (blockIdx.x == 0 && tid == 0) {
        for (int i = n4 * 4; i < n; ++i) ghmc_elem(x[i], y[i], mybins);
    }

    __syncthreads();

    // block reduce: 20 slots across 256 private regions, then one global atomic per slot
    if (tid < 2 * BINS) {
        if (tid < BINS) {
            unsigned a = 0u;
            for (int tt = 0; tt < TILE; ++tt) a += sHist[tt * STRIDE + tid];
            if (a) atomicAdd(&ws[tid], a);
        } else {
            float a = 0.0f;
            for (int tt = 0; tt < TILE; ++tt) a += __uint_as_float(sHist[tt * STRIDE + tid]);
            atomicAdd(((float*)ws) + tid, a);
        }
    }
}

__global__ void ghmc_final(const unsigned* __restrict__ ws, float* __restrict__ out) {
    if (threadIdx.x == 0) {
        const float* sums = ((const float*)ws) + BINS;
        int nb = 0;
#pragma unroll
        for (int b = 0; b < BINS; ++b) nb += (ws[b] != 0u) ? 1 : 0;
        float nn  = (nb > 0) ? (float)nb : 1.0f;
        float acc = 0.0f;
#pragma unroll
        for (int b = 0; b < BINS; ++b)
            if (ws[b] != 0u) acc += sums[b] / ((float)ws[b] * nn);
        out[0] = acc;   // == sum(w*bce)/t  (t cancels algebraically)
    }
}

extern "C" void kernel_launch(void* const* d_in, const int* in_sizes, int n_in,
                              void* d_out, int out_size, void* d_ws, size_t ws_size,
                              hipStream_t stream) {
    const float* x = (const float*)d_in[0];
    const float* y = (const float*)d_in[1];
    float*       out = (float*)d_out;
    const int n  = in_sizes[0];
    const int n4 = n >> 2;
    int tiles = (n4 + TILE - 1) / TILE;
    int grid  = tiles < 1 ? 1 : (tiles > MAXGRID ? MAXGRID : tiles);

    // ws[0..9] = u32 counts, ws[10..19] = f32 bin sums; must be zeroed every launch
    hipMemsetAsync(d_ws, 0, 2 * BINS * sizeof(unsigned), stream);
    ghmc_main<<<grid, TILE, 0, stream>>>(x, y, (unsigned*)d_ws, n, n4, tiles);
    ghmc_final<<<1, 32, 0, stream>>>((const unsigned*)d_ws, out);
}